// mHCConnection_56264071577889
// MI455X (gfx1250) — compile-verified
//
#include <hip/hip_runtime.h>
#include <math.h>

typedef float v2f __attribute__((ext_vector_type(2)));
typedef float v4f __attribute__((ext_vector_type(4)));
typedef float v8f __attribute__((ext_vector_type(8)));

#define NSTREAM 4
#define CH      1024
#define NC      4096      // NSTREAM * CH
#define TILE    16        // tokens per workgroup == WMMA M
#define WAVES   8
#define KPW     (NC / WAVES)   // 512 K per wave
#define EPSF    1e-8f

__global__ __launch_bounds__(256) void mhc_fused_kernel(
    const float* __restrict__ x,            // [B,S,n,C]
    const float* __restrict__ layer_output, // [B,S,C]
    const float* __restrict__ phi_post,     // [NC,4]
    const float* __restrict__ phi_res,      // [NC,16]
    const float* __restrict__ alpha_post_p, // [1]
    const float* __restrict__ alpha_res_p,  // [1]
    const float* __restrict__ b_post,       // [4]
    const float* __restrict__ b_res,        // [16] (eye(4) flattened)
    float* __restrict__ out)                // [B,S,n,C]
{
    __shared__ float red_res [WAVES][32][8];
    __shared__ float red_post[WAVES][32][8];
    __shared__ float ssql[WAVES][32];
    __shared__ float Gres [32][8];
    __shared__ float Gpost[32][8];
    __shared__ float sumsq[TILE];
    __shared__ float Hres [TILE][16];
    __shared__ float Hpost[TILE][4];

    const int t    = threadIdx.x;
    const int w    = t >> 5;          // wave id 0..7
    const int lane = t & 31;
    const int m    = lane & 15;       // token row for A frag / column for B frag
    const int kh   = lane >> 4;       // K-half within the 4-wide K step
    const int tile = blockIdx.x;

    const float* xt = x + (size_t)tile * TILE * NC;

    // warm layer_output tile in L2 (global_prefetch_b8)
    __builtin_prefetch(layer_output + (size_t)tile * TILE * CH + (size_t)t * 64, 0, 1);

    // ---------------- Stage 1: G = X_tile @ [phi_res | phi_post], ssq = rowwise sum(x^2)
    v8f acc_res  = {};
    v8f acc_post = {};
    float ssq = 0.0f;

    // phi_post has only 4 valid columns; lanes with m>=4 read a clamped (valid)
    // address and zero the value -> no exec-mask branch in the hot loop.
    const int   mpost = (m < NSTREAM) ? m : 0;
    const float msel  = (m < NSTREAM) ? 1.0f : 0.0f;

    const int kbase = w * KPW;
    for (int kk = 0; kk < KPW; kk += 4) {
        const int krow = kbase + kk + 2 * kh;           // this lane's K index pair base
        // A fragment: lane(0-15)=token m, VGPR0/1 = K krow,krow+1 ; lanes 16-31 carry K+2,K+3
        v2f a = *(const v2f*)(xt + (size_t)m * NC + krow);
        ssq += a.x * a.x + a.y * a.y;

        // B fragment for phi_res [K x 16]: lane's N = m, rows krow, krow+1 (64B coalesced / halfwave)
        v2f br;
        br.x = phi_res[krow * 16 + m];
        br.y = phi_res[(krow + 1) * 16 + m];
        acc_res = __builtin_amdgcn_wmma_f32_16x16x4_f32(
            false, a, false, br, (short)0, acc_res, false, false);

        // B fragment for phi_post [K x 4], columns 4..15 forced to zero (branch-free)
        v2f bp;
        bp.x = phi_post[krow * 4 + mpost] * msel;
        bp.y = phi_post[(krow + 1) * 4 + mpost] * msel;
        acc_post = __builtin_amdgcn_wmma_f32_16x16x4_f32(
            false, a, false, bp, (short)0, acc_post, false, false);
    }

    #pragma unroll
    for (int r = 0; r < 8; ++r) {
        red_res [w][lane][r] = acc_res[r];
        red_post[w][lane][r] = acc_post[r];
    }
    ssql[w][lane] = ssq;
    __syncthreads();

    // ---------------- Stage 2a: cross-wave reduction (256 threads = 32 lanes x 8 regs)
    {
        const int lr = t & 31, rr = t >> 5;
        float sr = 0.0f, sp = 0.0f;
        #pragma unroll
        for (int ww = 0; ww < WAVES; ++ww) {
            sr += red_res [ww][lr][rr];
            sp += red_post[ww][lr][rr];
        }
        Gres [lr][rr] = sr;
        Gpost[lr][rr] = sp;
    }
    if (t < TILE) {
        float s = 0.0f;
        #pragma unroll
        for (int ww = 0; ww < WAVES; ++ww) s += ssql[ww][t] + ssql[ww][t + 16];
        sumsq[t] = s;
    }
    __syncthreads();

    // ---------------- Stage 2b: per-token sigmoid gate + 4x4 Sinkhorn (16 threads)
    if (t < TILE) {
        const int   mm  = t;
        const float inv = rsqrtf(sumsq[mm] * (1.0f / NC) + EPSF); // 1/rms
        const float ap  = alpha_post_p[0];
        const float ar  = alpha_res_p[0];
        const int   hi  = (mm & 8) ? 16 : 0;  // C/D layout: lane = N + 16*(M>=8)
        const int   rg  = mm & 7;             //              vgpr = M & 7

        #pragma unroll
        for (int i = 0; i < 4; ++i) {
            float g  = Gpost[i + hi][rg];
            float ht = ap * g * inv + b_post[i];
            Hpost[mm][i] = 2.0f / (1.0f + expf(-ht));
        }

        float M4[4][4];
        #pragma unroll
        for (int i = 0; i < 4; ++i)
            #pragma unroll
            for (int j = 0; j < 4; ++j) {
                float g = Gres[(i * 4 + j) + hi][rg];
                M4[i][j] = expf(ar * g * inv + b_res[i * 4 + j]);
            }
        for (int it = 0; it < 20; ++it) {
            #pragma unroll
            for (int i = 0; i < 4; ++i) {
                float rs = M4[i][0] + M4[i][1] + M4[i][2] + M4[i][3] + EPSF;
                float rrv = 1.0f / rs;
                M4[i][0] *= rrv; M4[i][1] *= rrv; M4[i][2] *= rrv; M4[i][3] *= rrv;
            }
            #pragma unroll
            for (int j = 0; j < 4; ++j) {
                float cs = M4[0][j] + M4[1][j] + M4[2][j] + M4[3][j] + EPSF;
                float crv = 1.0f / cs;
                M4[0][j] *= crv; M4[1][j] *= crv; M4[2][j] *= crv; M4[3][j] *= crv;
            }
        }
        #pragma unroll
        for (int i = 0; i < 4; ++i)
            #pragma unroll
            for (int j = 0; j < 4; ++j)
                Hres[mm][i * 4 + j] = M4[i][j];
    }
    __syncthreads();

    // ---------------- Stage 3: out = H_res @ x + H_post (x) layer_output
    // x re-read is L2-hot; output stores are non-temporal (write-once stream,
    // keep L2 for x tiles).
    {
        const int mm  = t >> 4;     // token within tile
        const int j16 = t & 15;     // column group
        const float* xr = xt + (size_t)mm * NC;
        const float* lo = layer_output + ((size_t)tile * TILE + mm) * CH;
        float*       ot = out + (size_t)tile * TILE * NC + (size_t)mm * NC;

        float h[4][4], hp[4];
        #pragma unroll
        for (int i = 0; i < 4; ++i) {
            hp[i] = Hpost[mm][i];
            #pragma unroll
            for (int j = 0; j < 4; ++j) h[i][j] = Hres[mm][i * 4 + j];
        }

        for (int cc = j16 * 4; cc < CH; cc += 64) {
            v4f x0 = *(const v4f*)(xr + 0 * CH + cc);
            v4f x1 = *(const v4f*)(xr + 1 * CH + cc);
            v4f x2 = *(const v4f*)(xr + 2 * CH + cc);
            v4f x3 = *(const v4f*)(xr + 3 * CH + cc);
            v4f lv = *(const v4f*)(lo + cc);
            #pragma unroll
            for (int i = 0; i < 4; ++i) {
                v4f o = h[i][0] * x0 + h[i][1] * x1 + h[i][2] * x2 + h[i][3] * x3
                      + hp[i] * lv;
                __builtin_nontemporal_store(o, (v4f*)(ot + i * CH + cc));
            }
        }
    }
}

extern "C" void kernel_launch(void* const* d_in, const int* in_sizes, int n_in,
                              void* d_out, int out_size, void* d_ws, size_t ws_size,
                              hipStream_t stream) {
    (void)n_in; (void)out_size; (void)d_ws; (void)ws_size;
    // setup_inputs order:
    // 0:x 1:layer_output 2:phi_pre 3:phi_post 4:phi_res
    // 5:alpha_pre 6:alpha_post 7:alpha_res 8:b_pre 9:b_post 10:b_res
    const float* x            = (const float*)d_in[0];
    const float* layer_output = (const float*)d_in[1];
    const float* phi_post     = (const float*)d_in[3];
    const float* phi_res      = (const float*)d_in[4];
    const float* alpha_post   = (const float*)d_in[6];
    const float* alpha_res    = (const float*)d_in[7];
    const float* b_post       = (const float*)d_in[9];
    const float* b_res        = (const float*)d_in[10];
    float*       out          = (float*)d_out;

    const int tokens = in_sizes[0] / NC;      // B*S = 16384
    const int blocks = tokens / TILE;         // 1024

    mhc_fused_kernel<<<blocks, 256, 0, stream>>>(
        x, layer_output, phi_post, phi_res,
        alpha_post, alpha_res, b_post, b_res, out);
}